// MoRA_6485400617623
// MI455X (gfx1250) — compile-verified
//
#include <hip/hip_runtime.h>

// MoRA compress -> (rank x rank) GEMM -> repeat_interleave decompress.
// HBM-bound (536 MB traffic ~= 23 us floor @ 23.3 TB/s); GEMM done in full
// f32 precision with V_WMMA_F32_16X16X4_F32.

typedef __attribute__((ext_vector_type(2))) float v2f;
typedef __attribute__((ext_vector_type(4))) float v4f;
typedef __attribute__((ext_vector_type(8))) float v8f;

#define HIDDEN 4096
#define RANK   256
#define GROUPS 16            // HIDDEN / RANK
#define ROWS_PER_BLOCK 16
#define THREADS 256          // 8 wave32s
#define CS 260               // LDS row stride (floats); 260*4=1040B -> 16B aligned, bank-conflict-free
#define RS 260

__global__ __launch_bounds__(THREADS)
void mora_fused_kernel(const float* __restrict__ x,
                       const float* __restrict__ mat,
                       float* __restrict__ out,
                       int nRows) {
    __shared__ float comp[ROWS_PER_BLOCK * CS];   // compressed [16][256] (padded)
    __shared__ float res [ROWS_PER_BLOCK * RS];   // GEMM result [16][256] (padded)

    const int  tid  = threadIdx.x;
    const long row0 = (long)blockIdx.x * ROWS_PER_BLOCK;
    if (row0 >= nRows) return;

    // ---- Phase 1: compress. comp[r][j] = sum_c x[row0+r][c*RANK + j] ----
    // 256 threads = 4 rows x 64 float4-columns per pass; NT b128 loads, coalesced.
    {
        const int r_sub = tid >> 6;        // 0..3
        const int c4    = tid & 63;        // float4 column 0..63
        #pragma unroll
        for (int rr = 0; rr < ROWS_PER_BLOCK; rr += 4) {
            const int r = rr + r_sub;
            const v4f* xp = (const v4f*)(x + (row0 + r) * HIDDEN) + c4;
            v4f s = {0.f, 0.f, 0.f, 0.f};
            #pragma unroll
            for (int c = 0; c < GROUPS; ++c) {
                v4f t = __builtin_nontemporal_load(xp + c * (RANK / 4));
                s += t;
            }
            *(v4f*)&comp[r * CS + c4 * 4] = s;
        }
    }
    __syncthreads();

    // ---- Phase 2: res[16][256] = comp[16][256] @ mat[256][256] via f32 WMMA ----
    // Wave w owns N in [w*32, w*32+32) -> two 16x16 tiles; K loop = 64 steps of K=4.
    const int wave = tid >> 5;
    const int lane = tid & 31;
    const int lm   = lane & 15;            // M (for A) / N (for B) within tile
    const int hi   = lane >> 4;            // lane group
    const int koff = hi << 1;              // 0 or 2 (K sub-offset per ISA layout)
    const int n0   = wave * 32 + lm;

    v8f acc0 = {};
    v8f acc1 = {};
    const float* arow = comp + lm * CS + koff;   // even element index -> 8B aligned
    const float* b0p  = mat + n0;
    const float* b1p  = mat + n0 + 16;

    #pragma unroll 8
    for (int k = 0; k < RANK; k += 4) {
        v2f a = *(const v2f*)(arow + k);         // ds_load_b64, conflict-free (pad 260)
        v2f b0, b1;
        b0.x = b0p[(k + koff)     * RANK];
        b0.y = b0p[(k + koff + 1) * RANK];
        b1.x = b1p[(k + koff)     * RANK];
        b1.y = b1p[(k + koff + 1) * RANK];
        acc0 = __builtin_amdgcn_wmma_f32_16x16x4_f32(
                   false, a, false, b0, (short)0, acc0, false, false);
        acc1 = __builtin_amdgcn_wmma_f32_16x16x4_f32(
                   false, a, false, b1, (short)0, acc1, false, false);
    }

    // Park D fragments in LDS. 16x16 f32 C/D layout: lane L holds col N=L%16,
    // rows M = v + (L<16 ? 0 : 8) across the 8 acc VGPRs.
    #pragma unroll
    for (int v = 0; v < 8; ++v) {
        const int M = v + (hi << 3);
        res[M * RS + n0]      = acc0[v];
        res[M * RS + n0 + 16] = acc1[v];
    }
    __syncthreads();

    // ---- Phase 3: decompress (repeat_interleave x16), coalesced NT b128 stores ----
    // Row = 1024 float4 slots; slot p covers floats 4p..4p+3 -> all equal res[r][p>>2].
    #pragma unroll
    for (int r = 0; r < ROWS_PER_BLOCK; ++r) {
        float* op = out + (row0 + r) * HIDDEN;
        #pragma unroll
        for (int c = 0; c < 4; ++c) {
            const int p   = c * THREADS + tid;         // 0..1023
            const float v = res[r * RS + (p >> 2)];
            v4f v4 = {v, v, v, v};
            __builtin_nontemporal_store(v4, (v4f*)(op + p * 4));
        }
    }
}

extern "C" void kernel_launch(void* const* d_in, const int* in_sizes, int n_in,
                              void* d_out, int out_size, void* d_ws, size_t ws_size,
                              hipStream_t stream) {
    const float* x   = (const float*)d_in[0];   // [B, S, HIDDEN] f32
    const float* mat = (const float*)d_in[1];   // [RANK, RANK]  f32
    float* out = (float*)d_out;                 // [B, S, HIDDEN] f32

    const int nRows = in_sizes[0] / HIDDEN;                 // B*S = 16384
    const int grid  = (nRows + ROWS_PER_BLOCK - 1) / ROWS_PER_BLOCK;

    mora_fused_kernel<<<grid, THREADS, 0, stream>>>(x, mat, out, nRows);
}